// BiLSTM_CRF_11948599018244
// MI455X (gfx1250) — compile-verified
//
#include <hip/hip_runtime.h>

// Problem constants (match reference)
#define B_TOT     1024
#define LSEQ      1024
#define NTAG      32
#define START_TAG 30
#define STOP_TAG  31

typedef float v2f __attribute__((ext_vector_type(2)));
typedef float v8f __attribute__((ext_vector_type(8)));

// D = A(16x4 f32) x B(4x16 f32) + C(16x16 f32)   [V_WMMA_F32_16X16X4_F32]
__device__ __forceinline__ v8f wmma_f32_4(v2f a, v2f b, v8f c) {
  return __builtin_amdgcn_wmma_f32_16x16x4_f32(
      /*neg_a=*/false, a, /*neg_b=*/false, b,
      /*c_mod=*/(short)0, c, /*reuse_a=*/false, /*reuse_b=*/false);
}

// ---------------------------------------------------------------------------
// Kernel 1: gold path score.  out[b] = -(emit + transition score)
// ---------------------------------------------------------------------------
__global__ __launch_bounds__(32) void crf_gold_kernel(
    const float* __restrict__ feats, const float* __restrict__ trans,
    const int* __restrict__ tags, float* __restrict__ out) {
  const int b    = blockIdx.x;
  const int lane = threadIdx.x;
  const int*   tg = tags + (size_t)b * LSEQ;
  const float* fb = feats + (size_t)b * LSEQ * NTAG;

  float acc = 0.0f;
  for (int l = lane; l < LSEQ; l += 32) {
    const int cur  = tg[l];
    const int prev = (l == 0) ? START_TAG : tg[l - 1];
    acc += fb[(size_t)l * NTAG + cur] + trans[cur * NTAG + prev];
  }
  if (lane == 0) acc += trans[STOP_TAG * NTAG + tg[LSEQ - 1]];
#pragma unroll
  for (int o = 16; o > 0; o >>= 1) acc += __shfl_xor(acc, o, 32);
  if (lane == 0) out[b] = -acc;
}

// ---------------------------------------------------------------------------
// Kernel 2: forward algorithm (log-partition) in probability space.
//   p_t = diag(exp(feat_t)) * E * p_{t-1}   (rescaled every 4 steps)
//
// Row permutation sigma(n) = 4*((n&7)>>1) + (n&1) + 2*(n>>3) applied to E's
// rows makes the WMMA C-fragment output layout IDENTICAL to the next step's
// B-fragment layout: p stays in registers, the scan loop has no cross-lane
// ops and no LDS.  16 v_wmma_f32_16x16x4 per step as 8 chains of depth 2
// (occupancy is 1 wave/SIMD, so ILP is the only latency hiding).
// ---------------------------------------------------------------------------
__global__ __launch_bounds__(128, 1) void crf_forward_kernel(
    const float* __restrict__ feats, const float* __restrict__ trans,
    float* __restrict__ out) {
  __shared__ float Elds[NTAG * 33];   // exp(trans), padded stride 33
  __shared__ float Estop[NTAG];       // exp(trans[STOP][:])

  const int tid = threadIdx.x;
  for (int i = tid; i < NTAG * NTAG; i += 128) {
    const int r = i >> 5, c = i & 31;
    const float e = __expf(trans[i]);           // exp(-10000) == 0 -> masking
    Elds[r * 33 + c] = e;
    if (r == STOP_TAG) Estop[c] = e;
  }
  __syncthreads();

  const int wave = tid >> 5;
  const int lane = tid & 31;
  const int col  = lane & 15;     // M position within A-fragment / N column
  const int hi   = lane >> 4;
  const int b    = (blockIdx.x * 4 + wave) * 16 + col;

  // permuted row for this lane's A-fragment M position
  const int srow = 4 * ((col & 7) >> 1) + (col & 1) + 2 * (col >> 3);

  // A-fragments of E (rows permuted by sigma): tile0 rows sigma(0..15),
  // tile1 rows 16+sigma(0..15); 8 K-chunks of 4.
  v2f A0[8], A1[8];
#pragma unroll
  for (int k = 0; k < 8; ++k) {
    const int kk = 4 * k + 2 * hi;
    A0[k].x = Elds[srow * 33 + kk];
    A0[k].y = Elds[srow * 33 + kk + 1];
    A1[k].x = Elds[(16 + srow) * 33 + kk];
    A1[k].y = Elds[(16 + srow) * 33 + kk + 1];
  }

  // p held directly in B-fragment layout (== permuted C layout):
  // chunk k = regs (2k,2k+1): rows 4k/4k+1 (lanes<16), 4k+2/4k+3 (lanes>=16).
  float P0[8], P1[8];
#pragma unroll
  for (int v = 0; v < 8; ++v) { P0[v] = 0.0f; P1[v] = 0.0f; }
  if (hi) P1[6] = 1.0f;   // p0[START=30]: row 16+sigma(14)=30, lanes>=16

  float logC = 0.0f;
  // emission pairs: this lane needs rows (4m + 2*hi, +1) for m = 0..7
  const float* fbase = feats + (size_t)b * LSEQ * NTAG + 2 * hi;

  v2f cf[8];
#pragma unroll
  for (int m = 0; m < 8; ++m) cf[m] = *(const v2f*)(fbase + 4 * m);

  for (int t4 = 0; t4 < LSEQ / 4; ++t4) {
#pragma unroll
    for (int i = 0; i < 4; ++i) {
      const int t = 4 * t4 + i;

      // prefetch ~8 steps ahead (uniform clamped index, no branch)
      const int tp = (t + 8 < LSEQ) ? (t + 8) : (LSEQ - 1);
      __builtin_prefetch(fbase + (size_t)tp * NTAG, 0, 1);

      // issue next step's b64 loads early; scalar-clamped row index avoids
      // per-register select movs in the double buffer.
      const int tn = (t + 1 < LSEQ) ? (t + 1) : t;
      const float* pnext = fbase + (size_t)tn * NTAG;
      v2f nf[8];
#pragma unroll
      for (int m = 0; m < 8; ++m) nf[m] = *(const v2f*)(pnext + 4 * m);

      // B chunks come straight from P -- no layout conversion.
      v2f Bc[8];
#pragma unroll
      for (int k = 0; k < 4; ++k) {
        Bc[k].x     = P0[2 * k]; Bc[k].y     = P0[2 * k + 1];
        Bc[k + 4].x = P1[2 * k]; Bc[k + 4].y = P1[2 * k + 1];
      }

      // E @ p : 16 WMMAs as 8 independent chains of depth 2
      v8f c0[4], c1[4];
      const v8f z = {0.f,0.f,0.f,0.f,0.f,0.f,0.f,0.f};
#pragma unroll
      for (int m = 0; m < 4; ++m) {
        c0[m] = wmma_f32_4(A0[2 * m], Bc[2 * m], z);
        c1[m] = wmma_f32_4(A1[2 * m], Bc[2 * m], z);
      }
#pragma unroll
      for (int m = 0; m < 4; ++m) {
        c0[m] = wmma_f32_4(A0[2 * m + 1], Bc[2 * m + 1], c0[m]);
        c1[m] = wmma_f32_4(A1[2 * m + 1], Bc[2 * m + 1], c1[m]);
      }

      // p_new = (E p) * exp(feat); reg v of tile tau is row
      // tau*16 + 4*(v>>1) + (v&1) + 2*hi  ->  pair m = (v>>1) + 4*tau.
#pragma unroll
      for (int v = 0; v < 8; ++v) {
        const float f0 = (v & 1) ? cf[v >> 1].y : cf[v >> 1].x;
        const float f1 = (v & 1) ? cf[4 + (v >> 1)].y : cf[4 + (v >> 1)].x;
        const float s0 = (c0[0][v] + c0[1][v]) + (c0[2][v] + c0[3][v]);
        const float s1 = (c1[0][v] + c1[1][v]) + (c1[2][v] + c1[3][v]);
        P0[v] = s0 * __expf(f0);
        P1[v] = s1 * __expf(f1);
      }

#pragma unroll
      for (int m = 0; m < 8; ++m) cf[m] = nf[m];
    }

    // lazy rescaling every 4 steps (growth bounded well inside f32 range)
    float s = 0.0f;
#pragma unroll
    for (int v = 0; v < 8; ++v) s += P0[v] + P1[v];
    s += __shfl_xor(s, 16, 32);       // full 32-row column sum, all lanes
    const float rinv = 1.0f / s;
#pragma unroll
    for (int v = 0; v < 8; ++v) { P0[v] *= rinv; P1[v] *= rinv; }
    logC += __logf(s);
  }

  // logZ = log( sum_i p[i] * exp(trans[STOP][i]) ) + logC
  float fin = 0.0f;
#pragma unroll
  for (int v = 0; v < 8; ++v) {
    const int r = 4 * (v >> 1) + (v & 1) + 2 * hi;
    fin += P0[v] * Estop[r] + P1[v] * Estop[16 + r];
  }
  fin += __shfl_xor(fin, 16, 32);
  const float logZ = __logf(fin) + logC;
  if (lane < 16) out[b] += logZ;     // gold kernel already wrote -gold
}

// ---------------------------------------------------------------------------
extern "C" void kernel_launch(void* const* d_in, const int* in_sizes, int n_in,
                              void* d_out, int out_size, void* d_ws, size_t ws_size,
                              hipStream_t stream) {
  const float* feats = (const float*)d_in[0];   // [B, L, T] f32
  const float* trans = (const float*)d_in[1];   // [T, T]    f32
  const int*   tags  = (const int*)d_in[2];     // [B, L]    int32 (JAX x64 off)
  float*       out   = (float*)d_out;           // [B]       f32

  // 1) out[b] = -gold_score(b)
  crf_gold_kernel<<<B_TOT, 32, 0, stream>>>(feats, trans, tags, out);
  // 2) out[b] += logZ(b)   (16 columns per wave, 4 waves per block)
  crf_forward_kernel<<<B_TOT / 64, 128, 0, stream>>>(feats, trans, out);
}